// GraphEncoder_53214644797442
// MI455X (gfx1250) — compile-verified
//
#include <hip/hip_runtime.h>
#include <hip/hip_bf16.h>

#define IN_DIM   128
#define HID      64
#define OUT_DIM  128
#define NGRAPHS  128

typedef __attribute__((ext_vector_type(2))) float v2f;
typedef __attribute__((ext_vector_type(8))) float v8f;

// Native CDNA5 float atomic (VGLOBAL opcode 86, no-return form -> STOREcnt).
// Forced via inline asm so it can never fall back to a CAS loop.
static __device__ __forceinline__ void atomic_add_f32(float* p, float v) {
    asm volatile("global_atomic_add_f32 %0, %1, off"
                 :
                 : "v"(p), "v"(v)
                 : "memory");
}

// ---------------------------------------------------------------- utilities
__global__ void zero_kernel(float* __restrict__ p, int n) {
    int i = blockIdx.x * blockDim.x + threadIdx.x;
    if (i < n) p[i] = 0.0f;
}

// deg[dst] += 1 per edge
__global__ void edge_degree_kernel(const int* __restrict__ dst,
                                   float* __restrict__ deg, int E) {
    int e = blockIdx.x * blockDim.x + threadIdx.x;
    if (e < E) atomic_add_f32(&deg[dst[e]], 1.0f);
}

// dinv = rsqrt(deg+1), invdeg = 1/(deg+1)
__global__ void finalize_deg_kernel(const float* __restrict__ deg,
                                    float* __restrict__ dinv,
                                    float* __restrict__ invdeg, int n) {
    int i = blockIdx.x * blockDim.x + threadIdx.x;
    if (i < n) {
        float d = deg[i] + 1.0f;
        dinv[i]   = rsqrtf(d);
        invdeg[i] = 1.0f / d;
    }
}

// ------------------------------------------------- fp32 WMMA GEMM (C = A@B [+bias])
// One wave computes a 16-row x N_ stripe of C with NT=N_/16 accumulators, using
// V_WMMA_F32_16X16X4_F32.  A:[M,K] row-major, B:[K,N_] row-major, M%16==0, K%4==0.
// Per ISA 7.12.2 layouts:
//   A 16x4 : lanes 0-15 hold M=0..15 with {K=k0,k0+1}; lanes 16-31 hold {K=k0+2,k0+3}
//   B 4x16 : lane gives N=lane%16; VGPR0/1 hold the matching K rows
//   C/D    : VGPR v -> M=v (lanes 0-15) / M=v+8 (lanes 16-31), N=lane%16
template <int N_>
__global__ void wmma_gemm_rows_kernel(const float* __restrict__ A,
                                      const float* __restrict__ B,
                                      const float* __restrict__ bias,
                                      float* __restrict__ C,
                                      int M, int K) {
    constexpr int NT = N_ / 16;
    int wave = (int)((blockIdx.x * (unsigned)blockDim.x + threadIdx.x) >> 5);
    if (wave * 16 >= M) return;          // wave-uniform exit: EXEC stays all-ones
    int lane = threadIdx.x & 31;
    int half = lane >> 4;                // 0: lanes 0-15, 1: lanes 16-31
    int l15  = lane & 15;
    int row  = wave * 16 + l15;
    int koff = half * 2;

    v8f acc[NT];
#pragma unroll
    for (int t = 0; t < NT; ++t) acc[t] = (v8f){};

    const float* aptr = A + (size_t)row * K + koff;     // advances by +4 per step
    const float* bptr = B + (size_t)koff * N_ + l15;    // advances by +4*N_ per step

    for (int k0 = 0; k0 < K; k0 += 4) {
        v2f a;
        a.x = aptr[0];                   // contiguous pair -> single b64 load
        a.y = aptr[1];
#pragma unroll
        for (int t = 0; t < NT; ++t) {
            v2f b;
            b.x = bptr[t * 16];          // immediate offsets (N_ is constexpr)
            b.y = bptr[N_ + t * 16];
            acc[t] = __builtin_amdgcn_wmma_f32_16x16x4_f32(
                         /*neg_a=*/false, a, /*neg_b=*/false, b,
                         /*c_mod=*/(short)0, acc[t],
                         /*reuse_a=*/false, /*reuse_b=*/false);
        }
        aptr += 4;
        bptr += 4 * N_;
    }

    int r0 = wave * 16 + half * 8;
#pragma unroll
    for (int t = 0; t < NT; ++t) {
        int col = t * 16 + l15;
        float badd = bias ? bias[col] : 0.0f;
        float* cp = C + (size_t)r0 * N_ + col;
#pragma unroll
        for (int v = 0; v < 8; ++v) {
            cp[(size_t)v * N_] = acc[t][v] + badd;
        }
    }
}

// ----------------------------------------- edge scatter: agg[dst] += xw[src]*norm
// thread = (edge, 4-channel group); H == 64 fixed.
__global__ void edge_scatter_kernel(const float* __restrict__ xw,
                                    const int* __restrict__ src,
                                    const int* __restrict__ dst,
                                    const float* __restrict__ dinv,
                                    float* __restrict__ agg, int E) {
    long long tid = (long long)blockIdx.x * blockDim.x + threadIdx.x;
    long long e  = tid >> 4;
    int cg       = (int)(tid & 15);
    if (e >= E) return;
    int s = src[e], d = dst[e];
    float norm = dinv[s] * dinv[d];
    const float4* xp = reinterpret_cast<const float4*>(xw + (size_t)s * HID);
    float4 v = xp[cg];
    float* ap = agg + (size_t)d * HID + (cg << 2);
    atomic_add_f32(ap + 0, v.x * norm);
    atomic_add_f32(ap + 1, v.y * norm);
    atomic_add_f32(ap + 2, v.z * norm);
    atomic_add_f32(ap + 3, v.w * norm);
}

// h = relu(agg + xw*invdeg + bias), written in place over agg
__global__ void combine_relu_kernel(const float* __restrict__ xw,
                                    float* __restrict__ agg,
                                    const float* __restrict__ invdeg,
                                    const float* __restrict__ bias,
                                    int n_nodes) {
    int idx = blockIdx.x * blockDim.x + threadIdx.x;
    if (idx >= n_nodes * HID) return;
    int node = idx >> 6;
    int c    = idx & 63;
    float v = agg[idx] + xw[idx] * invdeg[node] + bias[c];
    agg[idx] = fmaxf(v, 0.0f);
}

// per-graph sums + counts
__global__ void pool_scatter_kernel(const float* __restrict__ h,
                                    const int* __restrict__ batch,
                                    float* __restrict__ gsum,
                                    float* __restrict__ gcnt, int n_nodes) {
    int idx = blockIdx.x * blockDim.x + threadIdx.x;
    if (idx >= n_nodes * HID) return;
    int node = idx >> 6;
    int c    = idx & 63;
    int g    = batch[node];
    atomic_add_f32(&gsum[g * HID + c], h[idx]);
    if (c == 0) atomic_add_f32(&gcnt[g], 1.0f);
}

__global__ void pool_div_kernel(const float* __restrict__ gsum,
                                const float* __restrict__ gcnt,
                                float* __restrict__ gpool) {
    int idx = blockIdx.x * blockDim.x + threadIdx.x;
    if (idx >= NGRAPHS * HID) return;
    int g = idx >> 6;
    gpool[idx] = gsum[idx] / fmaxf(gcnt[g], 1.0f);
}

// ---------------------------------------------------------------- launcher
template <int N_>
static inline void launch_gemm(const float* A, const float* B, const float* bias,
                               float* C, int M, int K, hipStream_t stream) {
    int waves  = M / 16;                   // M is a multiple of 16 for all calls
    int blocks = (waves + 7) / 8;          // 8 waves (256 threads) per block
    wmma_gemm_rows_kernel<N_><<<blocks, 256, 0, stream>>>(A, B, bias, C, M, K);
}

extern "C" void kernel_launch(void* const* d_in, const int* in_sizes, int n_in,
                              void* d_out, int out_size, void* d_ws, size_t ws_size,
                              hipStream_t stream) {
    const float* x     = (const float*)d_in[0];
    const int*   eidx  = (const int*)d_in[1];   // [2, E]: row0 = src, row1 = dst
    const int*   batch = (const int*)d_in[2];
    const float* W1    = (const float*)d_in[3];
    const float* b1    = (const float*)d_in[4];
    const float* W2    = (const float*)d_in[5];
    const float* b2    = (const float*)d_in[6];
    const float* Wl    = (const float*)d_in[7];
    const float* bl    = (const float*)d_in[8];
    float*       out   = (float*)d_out;

    const int n_nodes = in_sizes[0] / IN_DIM;   // 100000
    const int E       = in_sizes[1] / 2;        // 3200000
    const int* src = eidx;
    const int* dst = eidx + E;

    // workspace layout (floats)
    float* ws = (float*)d_ws;
    const size_t NF = (size_t)n_nodes * HID;    // 6.4M
    float* buf_xw  = ws;                        // [n_nodes, HID]
    float* buf_agg = buf_xw  + NF;              // [n_nodes, HID] (agg, then h in place)
    float* deg     = buf_agg + NF;              // [n_nodes]
    float* dinv    = deg     + n_nodes;
    float* invdeg  = dinv    + n_nodes;
    float* gsum    = invdeg  + n_nodes;         // [NGRAPHS, HID]
    float* gcnt    = gsum    + NGRAPHS * HID;   // [NGRAPHS]
    float* gpool   = gcnt    + NGRAPHS;         // [NGRAPHS, HID]

    const int T = 256;
    const int nodeGrid  = (n_nodes * HID + T - 1) / T;          // 25000
    const int edgeGrid  = (E + T - 1) / T;
    const long long sThreads = (long long)E * 16;
    const int scatGrid  = (int)((sThreads + T - 1) / T);        // 200000

    // degrees (shared by both layers)
    zero_kernel<<<(n_nodes + T - 1) / T, T, 0, stream>>>(deg, n_nodes);
    edge_degree_kernel<<<edgeGrid, T, 0, stream>>>(dst, deg, E);
    finalize_deg_kernel<<<(n_nodes + T - 1) / T, T, 0, stream>>>(deg, dinv, invdeg, n_nodes);

    // ---- layer 1
    launch_gemm<HID>(x, W1, nullptr, buf_xw, n_nodes, IN_DIM, stream);
    zero_kernel<<<nodeGrid, T, 0, stream>>>(buf_agg, n_nodes * HID);
    edge_scatter_kernel<<<scatGrid, T, 0, stream>>>(buf_xw, src, dst, dinv, buf_agg, E);
    combine_relu_kernel<<<nodeGrid, T, 0, stream>>>(buf_xw, buf_agg, invdeg, b1, n_nodes);
    // h1 now lives in buf_agg

    // ---- layer 2 (xw2 overwrites buf_xw; agg2 reuses buf_agg after GEMM consumes h1)
    launch_gemm<HID>(buf_agg, W2, nullptr, buf_xw, n_nodes, HID, stream);
    zero_kernel<<<nodeGrid, T, 0, stream>>>(buf_agg, n_nodes * HID);
    edge_scatter_kernel<<<scatGrid, T, 0, stream>>>(buf_xw, src, dst, dinv, buf_agg, E);
    combine_relu_kernel<<<nodeGrid, T, 0, stream>>>(buf_xw, buf_agg, invdeg, b2, n_nodes);
    // h2 now lives in buf_agg

    // ---- global mean pool
    zero_kernel<<<(NGRAPHS * HID + NGRAPHS + T - 1) / T, T, 0, stream>>>(gsum, NGRAPHS * HID + NGRAPHS);
    pool_scatter_kernel<<<nodeGrid, T, 0, stream>>>(buf_agg, batch, gsum, gcnt, n_nodes);
    pool_div_kernel<<<(NGRAPHS * HID + T - 1) / T, T, 0, stream>>>(gsum, gcnt, gpool);

    // ---- head: out = gpool @ Wl + bl
    launch_gemm<OUT_DIM>(gpool, Wl, bl, out, NGRAPHS, HID, stream);
}